// GIN_Node_44908178047327
// MI455X (gfx1250) — compile-verified
//
#include <hip/hip_runtime.h>
#include <hip/hip_bf16.h>

// GIN node classifier for MI455X (gfx1250).
// Bottleneck: edge gather/scatter moves ~2.5-3 GB/call -> ~120us at 23.3 TB/s.
// GEMMs (~10 GFLOP) run as fp32 WMMA (V_WMMA_F32_16X16X4_F32), weights staged
// transposed+padded in LDS (conflict-free ds_load_b64 fragments). A sched
// barrier separates the DS-load burst from the WMMA burst so in-order DS
// returns are hidden behind the matrix pipe (staggered dscnt waits) instead
// of a full s_wait_dscnt 0 per fragment.

typedef float v2f __attribute__((ext_vector_type(2)));
typedef float v8f __attribute__((ext_vector_type(8)));

#define N_NODES 50000
#define D_CH    128
#define OUT_CH  40
#define N_EDGES 600000
#define N_LAYERS 3

// ---------------- zero workspace (graph-capture safe memset) ----------------
__global__ void zero_f4_kernel(float4* __restrict__ p, long long n4) {
    long long i = (long long)blockIdx.x * blockDim.x + threadIdx.x;
    if (i < n4) p[i] = make_float4(0.f, 0.f, 0.f, 0.f);
}

// ---------------- edge gather + scatter-add ----------------
// One wave per edge: 32 lanes x float4 = 128 channels.
// global_load_b128 gather, 4x global_atomic_add_f32 scatter.
__global__ void gin_scatter_kernel(const float* __restrict__ h,
                                   const int* __restrict__ edge_index, // [2, E]
                                   float* __restrict__ aggr,
                                   int nEdges) {
    int wavesPerBlock = blockDim.x >> 5;
    int e = blockIdx.x * wavesPerBlock + (threadIdx.x >> 5);
    if (e >= nEdges) return;
    int lane = threadIdx.x & 31;
    int src = edge_index[e];
    int dst = edge_index[nEdges + e];
    const float4 v = *(const float4*)(h + (long long)src * D_CH + lane * 4);
    float* a = aggr + (long long)dst * D_CH + lane * 4;
    atomicAdd(a + 0, v.x);
    atomicAdd(a + 1, v.y);
    atomicAdd(a + 2, v.z);
    atomicAdd(a + 3, v.w);
}

// ---------------- WMMA fp32 GEMM: out = act(A(+A2) @ W + bias) ----------------
// One wave computes a 16-row x (NT*16)-col block; K = 128.
// W staged transposed in LDS: Wt[n][k] = W[k][n], row stride 132 floats ->
//   b fragment (W[ka][n], W[ka+1][n]) is one 8B-aligned ds_load_b64 with
//   bank = (4n + ka + {0..3}) mod 64 distinct across the wave (conflict-free).
// A 16x4 f32 layout: lane l (l16=l&15, half=l>>4): M=l16,
//   vgpr0 = A[M][k+2*half], vgpr1 = A[M][k+2*half+1]. B mirrors with same ka.
// C/D 16x16 f32: acc[v] holds (M = v + 8*half, N = l16).
template <int NT, bool RELU, bool ADDIN>
__global__ void gin_gemm_kernel(const float* __restrict__ A,
                                const float* __restrict__ A2,
                                const float* __restrict__ W,
                                const float* __restrict__ bias,
                                float* __restrict__ out,
                                int nrows, int ncols) {
    constexpr int NT16 = NT * 16;
    constexpr int LS = 132;                    // padded LDS row stride (floats)
    __shared__ __attribute__((aligned(16))) float Wt[NT16 * LS];

    // Cooperative stage of W^T (zero-fill ragged classifier columns).
    for (int idx = threadIdx.x; idx < D_CH * NT16; idx += blockDim.x) {
        int k = idx / NT16;
        int n = idx - k * NT16;
        float w = (n < ncols) ? W[(long long)k * ncols + n] : 0.f;
        Wt[n * LS + k] = w;
    }
    __syncthreads();

    const int wavesPerBlock = blockDim.x >> 5;
    int wave = blockIdx.x * wavesPerBlock + (threadIdx.x >> 5);
    int rb = wave * 16;
    if (rb >= nrows) return;                   // wave-uniform: EXEC all-1s below

    int lane = threadIdx.x & 31;
    int half = lane >> 4;                      // 0 or 1
    int l16  = lane & 15;

    const float* arow  = A + (long long)(rb + l16) * D_CH;
    const float* arow2 = ADDIN ? (A2 + (long long)(rb + l16) * D_CH) : nullptr;
    const float* lwt   = &Wt[l16 * LS];        // per-lane LDS base; tile/k fold to imms

    v8f acc[NT];
#pragma unroll
    for (int t = 0; t < NT; ++t) acc[t] = v8f{0.f, 0.f, 0.f, 0.f, 0.f, 0.f, 0.f, 0.f};

#pragma unroll 2
    for (int k = 0; k < D_CH; k += 4) {
        int ka = k + 2 * half;
        v2f a;
        a.x = arow[ka];
        a.y = arow[ka + 1];
        if (ADDIN) {
            a.x += arow2[ka];
            a.y += arow2[ka + 1];
        }
        // Phase 1: issue every B-fragment DS load for this k-step.
        v2f b[NT];
#pragma unroll
        for (int t = 0; t < NT; ++t)
            b[t] = *(const v2f*)(lwt + t * 16 * LS + ka);     // ds_load_b64
        // Keep the loads batched: scheduler may not sink them past this point,
        // so the in-order DS returns drain under the WMMA burst below.
        __builtin_amdgcn_sched_barrier(0);
        // Phase 2: WMMA burst.
#pragma unroll
        for (int t = 0; t < NT; ++t)
            acc[t] = __builtin_amdgcn_wmma_f32_16x16x4_f32(
                false, a, false, b[t], (short)0, acc[t], false, false);
    }

    // fused bias + activation + store (divergence only after all WMMAs)
#pragma unroll
    for (int t = 0; t < NT; ++t) {
        int n = t * 16 + l16;
        if (n >= ncols) continue;
        float bv = bias[n];
#pragma unroll
        for (int v = 0; v < 8; ++v) {
            int r = rb + v + 8 * half;
            float val = acc[t][v] + bv;
            if (RELU) val = fmaxf(val, 0.f);
            out[(long long)r * ncols + n] = val;
        }
    }
}

extern "C" void kernel_launch(void* const* d_in, const int* in_sizes, int n_in,
                              void* d_out, int out_size, void* d_ws, size_t ws_size,
                              hipStream_t stream) {
    const float* x  = (const float*)d_in[0];
    // d_in[1] = edge_attr: unused by the reference computation
    const int*   ei = (const int*)d_in[2];       // [2, E] (int indices)
    const float* W1 = (const float*)d_in[3];     // [L, 128, 128]
    const float* b1 = (const float*)d_in[4];     // [L, 128]
    const float* W2 = (const float*)d_in[5];     // [L, 128, 128]
    const float* b2 = (const float*)d_in[6];     // [L, 128]
    const float* Wc = (const float*)d_in[7];     // [128, 40]
    const float* bc = (const float*)d_in[8];     // [40]
    float* out = (float*)d_out;

    const long long nd = (long long)N_NODES * D_CH;   // 6.4M floats per buffer
    float* aggr = (float*)d_ws;                       // [N,128]
    float* hid  = aggr + nd;                          // [N,128]
    float* hA   = hid + nd;                           // [N,128] ping
    float* hB   = hA + nd;                            // [N,128] pong

    dim3 blk(256);
    int zeroGrid = (int)((nd / 4 + 255) / 256);
    int scGrid   = (N_EDGES + 7) / 8;                 // 8 waves/block, 1 edge/wave
    int gmGrid   = (N_NODES / 16 + 7) / 8;            // 8 waves/block, 16 rows/wave

    const float* hin = x;
    float* houts[N_LAYERS] = {hA, hB, hA};
    for (int i = 0; i < N_LAYERS; ++i) {
        zero_f4_kernel<<<zeroGrid, blk, 0, stream>>>((float4*)aggr, nd / 4);
        gin_scatter_kernel<<<scGrid, blk, 0, stream>>>(hin, ei, aggr, N_EDGES);
        // hid = relu((hin + aggr) @ W1[i] + b1[i])
        gin_gemm_kernel<8, true, true><<<gmGrid, blk, 0, stream>>>(
            hin, aggr, W1 + (long long)i * D_CH * D_CH, b1 + i * D_CH, hid, N_NODES, D_CH);
        // h = relu(hid @ W2[i] + b2[i])
        gin_gemm_kernel<8, true, false><<<gmGrid, blk, 0, stream>>>(
            hid, nullptr, W2 + (long long)i * D_CH * D_CH, b2 + i * D_CH, houts[i], N_NODES, D_CH);
        hin = houts[i];
    }
    // out = h @ Wc + bc   (40 cols -> 3 ragged 16-col tiles)
    gin_gemm_kernel<3, false, false><<<gmGrid, blk, 0, stream>>>(
        hin, nullptr, Wc, bc, out, N_NODES, OUT_CH);
}